// Projection_58025008169122
// MI455X (gfx1250) — compile-verified
//
#include <hip/hip_runtime.h>
#include <math.h>

// ---------------------------------------------------------------------------
// PET forward projection for MI455X (gfx1250).
// 3 axes x 200k LORs x 128 slices x 5x5 Gaussian footprint.
// x/y axes: double-buffered async stage of each 128x128 slice (64KB) into LDS
//   with CDNA5 global_load_async_to_lds_b128; s_wait_asynccnt 0x10 overlaps
//   slice k+1's DMA with slice k's compute. Gather from LDS.
// z axis: slice is stride-128 in memory -> gather directly from the
//   L2-resident 8MB image instead of an uncoalesced stage.
// Hot loop is division-free and works in cell units (DXYZ folded into the
// Gaussian constant); cutoff test hoisted per row.
// ---------------------------------------------------------------------------

constexpr int    G     = 128;
constexpr float  DXYZ  = 3.125f;                 // 400/128
constexpr float  INV_D = 0.32f;                  // 1/3.125
constexpr float  MINC  = -200.0f;                // center - size/2
constexpr double PI_D  = 3.14159265358979323846;
// exp(-d2_phys/(2s^2)) = exp2(-C2 * d2_cell),  d2_phys = d2_cell*DXYZ^2, 2s^2 = 4.5*pi
constexpr float  NC2   = (float)(-(3.125 * 3.125) / (4.5 * PI_D) * 1.4426950408889634);
constexpr float  KW2C  = (float)((9.0 * PI_D) / (3.125 * 3.125));  // cutoff in cell^2

__device__ __forceinline__ float fast_exp2(float x) {
#if __has_builtin(__builtin_amdgcn_exp2f)
    return __builtin_amdgcn_exp2f(x);   // v_exp_f32
#else
    return exp2f(x);
#endif
}

template <int AXIS>
__global__ __launch_bounds__(256) void proj_kernel(const float* __restrict__ image,
                                                   const float* __restrict__ lors,
                                                   float* __restrict__ out, int n)
{
    // ping-pong slice buffers (x/y axes only): 2 x 64KB of the 320KB WGP LDS
    __shared__ float sl[(AXIS == 2) ? 4 : (2 * G * G)];

    const int tid = threadIdx.x;
    const int gid = blockIdx.x * 256 + tid;
    const int idx = (gid < n) ? gid : (n - 1);   // clamp; still help stage slices

    // ---- load LOR endpoints, permuted so component 2 is the march axis ----
    const float* l = lors + (size_t)idx * 6;
    const float a0 = l[0], a1 = l[1], a2 = l[2], a3 = l[3], a4 = l[4], a5 = l[5];
    float p0x, p0y, p0z, p1x, p1y, p1z;
    if (AXIS == 0)      { p0x = a1; p0y = a2; p0z = a0; p1x = a4; p1y = a5; p1z = a3; }
    else if (AXIS == 1) { p0x = a0; p0y = a2; p0z = a1; p1x = a3; p1y = a5; p1z = a4; }
    else                { p0x = a0; p0y = a1; p0z = a2; p1x = a3; p1y = a4; p1z = a5; }

    const float dvx = p1x - p0x, dvy = p1y - p0y, dvz = p1z - p0z;
    const float L   = sqrtf(dvx * dvx + dvy * dvy + dvz * dvz);
    const float inv_dzl = 1.0f / dvz;             // only real division, per-LOR
    const float path = DXYZ * L * fabsf(inv_dzl);

    // ---- everything along the ray is affine in k: precompute slopes ----
    const float t0  = (MINC + 0.5f * DXYZ - p0z) * inv_dzl;  // t at k=0
    const float ts  = DXYZ * inv_dzl;                        // dt/dk
    const float fx0 = (p0x + t0 * dvx - MINC) * INV_D;       // cell-space x at k=0
    const float fxs = ts * dvx * INV_D;                      // dfx/dk
    const float fy0 = (p0y + t0 * dvy - MINC) * INV_D;
    const float fys = ts * dvy * INV_D;

    uint32_t lds_base = 0;
    if (AXIS != 2) {
        lds_base = (uint32_t)(uintptr_t)(&sl[0]);  // flat->LDS offset (low 32b)
        // prologue: async-stage slice 0 into buffer 0
#pragma unroll
        for (int it = 0; it < 16; ++it) {
            const int e4 = it * 256 + tid;
            const uint32_t loff = lds_base + (uint32_t)e4 * 16u;
            uint32_t goff;
            if (AXIS == 0) goff = (uint32_t)(e4 * 16);
            else {
                const int ii = e4 >> 5;
                goff = (uint32_t)(ii * 65536 + (e4 & 31) * 16);
            }
            asm volatile("global_load_async_to_lds_b128 %0, %1, %2"
                         :: "v"(loff), "v"(goff), "s"(image) : "memory");
        }
    }

    float acc = 0.0f;

    for (int k = 0; k < G; ++k) {
        const float* slice = sl + (k & 1) * (G * G);   // compute buffer (x/y)

        if (AXIS != 2) {
            // ---- issue slice k+1 (clamped) into the other buffer ----
            const int kk = (k + 1 < G) ? (k + 1) : (G - 1);
            const uint32_t bufoff = (uint32_t)(((k + 1) & 1) * 65536);
#pragma unroll
            for (int it = 0; it < 16; ++it) {
                const int e4 = it * 256 + tid;                 // 16B chunk index
                const uint32_t loff = lds_base + bufoff + (uint32_t)e4 * 16u;
                uint32_t goff;
                if (AXIS == 0) {
                    // image[k, ii, jj]: contiguous 64KB at kk*16384 floats
                    goff = (uint32_t)(kk * 65536 + e4 * 16);
                } else {
                    // image[ii, k, jj]: 128 rows of 512B, row stride 64KB
                    const int ii = e4 >> 5;
                    goff = (uint32_t)(ii * 65536 + kk * 512 + (e4 & 31) * 16);
                }
                asm volatile("global_load_async_to_lds_b128 %0, %1, %2"
                             :: "v"(loff), "v"(goff), "s"(image) : "memory");
            }
            // wait for slice k's 16 loads (the 16 just issued stay in flight)
            asm volatile("s_wait_asynccnt 0x10" ::: "memory");
            __syncthreads();  // all waves' slice-k data visible in LDS
        }

        // ---- ray/slice intersection (3 FMAs, no division) ----
        const float kf = (float)k;
        const float t  = fmaf(kf, ts, t0);
        const float pv = (t >= 0.0f && t <= 1.0f) ? path : 0.0f;
        const float fx = fmaf(kf, fxs, fx0);
        const float fy = fmaf(kf, fys, fy0);
        const float fi = floorf(fx);
        const float fj = floorf(fy);
        const int i0 = (int)fi;
        const int j0 = (int)fj;
        const float frx = fi - fx;    // -frac(fx), in (-1, 0]
        const float fry = fj - fy;

        // ---- separable Gaussian in cell units: 5 row + 5 col weights ----
        float wx[5], wy[5], tha[5], dy2[5];
        int ic[5], jc[5];
#pragma unroll
        for (int a = 0; a < 5; ++a) {
            const int ii = i0 + a - 2;
            const float du = frx + ((float)a - 1.5f);
            const float d2 = du * du;
            const float ex = fast_exp2(NC2 * d2);
            wx[a]  = ((unsigned)ii < (unsigned)G) ? ex : 0.0f;  // bounds into weight
            tha[a] = KW2C - d2;                                  // row cutoff threshold
            ic[a]  = ii < 0 ? 0 : (ii > G - 1 ? G - 1 : ii);

            const int jj = j0 + a - 2;
            const float dv = fry + ((float)a - 1.5f);
            const float e2 = dv * dv;
            const float ey = fast_exp2(NC2 * e2);
            wy[a] = ((unsigned)jj < (unsigned)G) ? ey : 0.0f;
            dy2[a] = e2;
            jc[a]  = jj < 0 ? 0 : (jj > G - 1 ? G - 1 : jj);
        }

        float sum = 0.0f;
#pragma unroll
        for (int a = 0; a < 5; ++a) {
            const int rowbase = ic[a] * G;
            const float* rp = (AXIS == 2) ? (image + ic[a] * 16384 + k) : nullptr;
            float rowsum = 0.0f;
#pragma unroll
            for (int b = 0; b < 5; ++b) {
                // circular cutoff: dx2[a] + dy2[b] <= KW2  <=>  dy2[b] <= tha[a]
                const float wyb = (dy2[b] <= tha[a]) ? wy[b] : 0.0f;
                const float v = (AXIS != 2) ? slice[rowbase + jc[b]]  // LDS gather
                                            : rp[jc[b] * 128];        // L2 gather
                rowsum = fmaf(wyb, v, rowsum);
            }
            sum = fmaf(wx[a], rowsum, sum);
        }
        acc = fmaf(sum, pv, acc);

        if (AXIS != 2) __syncthreads();  // slice k consumed before next overwrite
    }

    if (gid < n) out[gid] = acc;
}

extern "C" void kernel_launch(void* const* d_in, const int* in_sizes, int n_in,
                              void* d_out, int out_size, void* d_ws, size_t ws_size,
                              hipStream_t stream)
{
    const float* image = (const float*)d_in[0];
    const float* xlors = (const float*)d_in[1];
    const float* ylors = (const float*)d_in[2];
    const float* zlors = (const float*)d_in[3];
    float* out = (float*)d_out;

    const int n = in_sizes[1] / 6;           // N_LORS
    const int blocks = (n + 255) / 256;

    proj_kernel<0><<<blocks, 256, 0, stream>>>(image, xlors, out + 0 * (size_t)n, n);
    proj_kernel<1><<<blocks, 256, 0, stream>>>(image, ylors, out + 1 * (size_t)n, n);
    proj_kernel<2><<<blocks, 256, 0, stream>>>(image, zlors, out + 2 * (size_t)n, n);
}